// NCELoss_74887049773782
// MI455X (gfx1250) — compile-verified
//
#include <hip/hip_runtime.h>

// ---------------------------------------------------------------------------
// NCE loss (with faithful "shared negative pool" bug) for MI455X / gfx1250.
//   Pass 1: stream v,a,t once (201 MB -> ~8.6us @ 23.3 TB/s floor).
//           Per 16-row tile: V_WMMA_F32_16X16X4_F32 Gram blocks for diag dots
//           (v.a, v.t), VALU FMAs for row norms and v0-row sims, all from the
//           same WMMA operand registers (no extra traffic).
//   Pass 2: single block, O(N) epilogue -> two scalar losses.
// ---------------------------------------------------------------------------

#define N_ROWS 16384
#define DIM    1024
#define TEMP   0.07f
#define EPSN   1e-12f

typedef float v2f __attribute__((ext_vector_type(2)));
typedef float v8f __attribute__((ext_vector_type(8)));

// Capability check must only run on the DEVICE pass; on the host pass the
// AMDGCN builtins are (correctly) absent.
#if defined(__HIP_DEVICE_COMPILE__) && !__has_builtin(__builtin_amdgcn_wmma_f32_16x16x4_f32)
#define USE_ASM_WMMA 1
#else
#define USE_ASM_WMMA 0
#endif

__device__ __forceinline__ v8f wmma_16x16x4_f32(v2f a, v2f b, v8f c) {
#if USE_ASM_WMMA
  // Fallback: emit the ISA op directly; tie D to C (accumulate in place).
  asm volatile("v_wmma_f32_16x16x4_f32 %0, %1, %2, %0"
               : "+v"(c)
               : "v"(a), "v"(b));
  return c;
#else
  // 8-arg form: (neg_a, A, neg_b, B, c_mod, C, reuse_a, reuse_b)
  return __builtin_amdgcn_wmma_f32_16x16x4_f32(
      false, a, false, b, (short)0, c, false, false);
#endif
}

__device__ __forceinline__ float v8_sel(v8f x, int i) {
  float r = x[0];
  r = (i == 1) ? x[1] : r;
  r = (i == 2) ? x[2] : r;
  r = (i == 3) ? x[3] : r;
  r = (i == 4) ? x[4] : r;
  r = (i == 5) ? x[5] : r;
  r = (i == 6) ? x[6] : r;
  r = (i == 7) ? x[7] : r;
  return r;
}

// ---------------------------------------------------------------------------
// Pass 1: 256 threads = 8 waves = 4 tiles x 2 K-halves.
// Workspace layout (float[7][N]): nv2, na2, nt2, dva, dvt, r0a, r0t.
// ---------------------------------------------------------------------------
__global__ __launch_bounds__(256) void nce_pass1(
    const float* __restrict__ v, const float* __restrict__ a,
    const float* __restrict__ t, float* __restrict__ ws) {
  __shared__ float part[8][7][16];

  const int lane  = threadIdx.x & 31;
  const int wave  = threadIdx.x >> 5;
  const int tib   = wave >> 1;              // tile within block (0..3)
  const int khalf = wave & 1;               // which K half this wave owns
  const int r0    = (blockIdx.x * 4 + tib) * 16;
  const int m     = lane & 15;              // row (A) / col (B) index in tile
  const int kp    = (lane >> 4) * 2;        // K sub-offset per f32 WMMA layout
  const int k0    = khalf * (DIM / 2);
  const int k1    = k0 + (DIM / 2);

  const float* vp  = v + (size_t)(r0 + m) * DIM + kp;  // A fragment source
  const float* ap  = a + (size_t)(r0 + m) * DIM + kp;  // B fragment source
  const float* tp  = t + (size_t)(r0 + m) * DIM + kp;  // B fragment source
  const float* z0p = v + kp;                           // v row 0 (broadcast)

  v8f acc_a = {};                          // Gram block V * A^T (16x16)
  v8f acc_t = {};                          // Gram block V * T^T (16x16)
  float nv = 0.f, na = 0.f, nt = 0.f, ra = 0.f, rt = 0.f;

#pragma unroll 4
  for (int k = k0; k < k1; k += 4) {
    v2f av = *(const v2f*)(vp + k);        // A: lane holds (m, k+kp+{0,1})
    v2f ba = *(const v2f*)(ap + k);        // B: lane holds (k+kp+{0,1}, m)
    v2f bt = *(const v2f*)(tp + k);
    v2f z0 = *(const v2f*)(z0p + k);       // v0[k+kp+{0,1}] (lane-broadcast)

    acc_a = wmma_16x16x4_f32(av, ba, acc_a);
    acc_t = wmma_16x16x4_f32(av, bt, acc_t);

    // Free riders on the already-loaded operands:
    nv = fmaf(av.x, av.x, fmaf(av.y, av.y, nv));   // ||v_row||^2 partial
    na = fmaf(ba.x, ba.x, fmaf(ba.y, ba.y, na));   // ||a_row||^2 partial
    nt = fmaf(bt.x, bt.x, fmaf(bt.y, bt.y, nt));   // ||t_row||^2 partial
    ra = fmaf(z0.x, ba.x, fmaf(z0.y, ba.y, ra));   // v0 . a_row partial
    rt = fmaf(z0.x, bt.x, fmaf(z0.y, bt.y, rt));   // v0 . t_row partial
  }

  // Lane m and lane m+16 each saw half of row m's K elements: combine.
  nv += __shfl_xor(nv, 16, 32);
  na += __shfl_xor(na, 16, 32);
  nt += __shfl_xor(nt, 16, 32);
  ra += __shfl_xor(ra, 16, 32);
  rt += __shfl_xor(rt, 16, 32);

  // Diagonal of the 16x16 f32 C/D layout:
  //   m<8  -> acc[m]   @ lane m
  //   m>=8 -> acc[m-8] @ lane m+16
  float ca = v8_sel(acc_a, lane & 7);
  float ct = v8_sel(acc_t, lane & 7);
  const int srcl = (m < 8) ? m : (m + 16);
  float da = __shfl(ca, srcl, 32);
  float dt = __shfl(ct, srcl, 32);

  if (lane < 16) {
    part[wave][0][lane] = nv;
    part[wave][1][lane] = na;
    part[wave][2][lane] = nt;
    part[wave][3][lane] = da;
    part[wave][4][lane] = dt;
    part[wave][5][lane] = ra;
    part[wave][6][lane] = rt;
  }
  __syncthreads();

  if (khalf == 0 && lane < 16) {
    const int row = r0 + lane;
#pragma unroll
    for (int q = 0; q < 7; ++q) {
      float s = part[wave][q][lane] + part[wave + 1][q][lane];
      ws[(size_t)q * N_ROWS + row] = s;
    }
  }
}

// ---------------------------------------------------------------------------
// Pass 2: one 1024-thread block; O(N) epilogue.
// ---------------------------------------------------------------------------
__device__ __forceinline__ float blk_reduce(float x, float* sbuf) {
  const int tid = threadIdx.x;
  __syncthreads();
  sbuf[tid] = x;
  __syncthreads();
  for (int s = 512; s > 0; s >>= 1) {
    if (tid < s) sbuf[tid] += sbuf[tid + s];
    __syncthreads();
  }
  return sbuf[0];
}

__device__ __forceinline__ float rn(float sq) {
  return fmaxf(sqrtf(sq), EPSN);
}

__global__ __launch_bounds__(1024) void nce_pass2(
    const float* __restrict__ v, const float* __restrict__ a,
    const float* __restrict__ t, const float* __restrict__ ws,
    float* __restrict__ out) {
  __shared__ float sbuf[1024];
  const int tid = threadIdx.x;

  // extra = v1 . a0 (and v1 . t0), raw dots.
  float pa = 0.f, pt = 0.f;
  for (int i = tid; i < DIM; i += 1024) {
    float v1 = v[DIM + i];
    pa = fmaf(v1, a[i], pa);
    pt = fmaf(v1, t[i], pt);
  }
  const float ea_raw = blk_reduce(pa, sbuf);
  const float et_raw = blk_reduce(pt, sbuf);

  const float* nv2 = ws;
  const float* na2 = ws + 1 * (size_t)N_ROWS;
  const float* nt2 = ws + 2 * (size_t)N_ROWS;
  const float* dva = ws + 3 * (size_t)N_ROWS;
  const float* dvt = ws + 4 * (size_t)N_ROWS;
  const float* r0a = ws + 5 * (size_t)N_ROWS;
  const float* r0t = ws + 6 * (size_t)N_ROWS;

  const float sv0 = rn(nv2[0]);
  const float sv1 = rn(nv2[1]);
  const float sa0 = rn(na2[0]);
  const float st0 = rn(nt2[0]);

  // sum_neg = sum_{j=1..N-1} exp(sim(v0, x_j)/T) + exp(sim(v1, x_0)/T)
  float la = 0.f, lt = 0.f;
  for (int j = tid; j < N_ROWS; j += 1024) {
    if (j >= 1) {
      la += expf(r0a[j] / (sv0 * rn(na2[j])) / TEMP);
      lt += expf(r0t[j] / (sv0 * rn(nt2[j])) / TEMP);
    }
  }
  const float sna = blk_reduce(la, sbuf) + expf(ea_raw / (sv1 * sa0) / TEMP);
  const float snt = blk_reduce(lt, sbuf) + expf(et_raw / (sv1 * st0) / TEMP);

  // loss = mean_k [ log(exp(pos_k) + sum_neg) - pos_k ], NaN -> 0
  float acc_a = 0.f, acc_t = 0.f;
  for (int k = tid; k < N_ROWS; k += 1024) {
    const float svk = rn(nv2[k]);
    const float pos_a = dva[k] / (svk * rn(na2[k])) / TEMP;
    const float pos_t = dvt[k] / (svk * rn(nt2[k])) / TEMP;
    float loa = logf(expf(pos_a) + sna) - pos_a;
    float lot = logf(expf(pos_t) + snt) - pos_t;
    loa = (loa != loa) ? 0.f : loa;
    lot = (lot != lot) ? 0.f : lot;
    acc_a += loa;
    acc_t += lot;
  }
  const float suma = blk_reduce(acc_a, sbuf);
  const float sumt = blk_reduce(acc_t, sbuf);

  if (tid == 0) {
    out[0] = suma / (float)N_ROWS;
    out[1] = sumt / (float)N_ROWS;
  }
}

// ---------------------------------------------------------------------------
extern "C" void kernel_launch(void* const* d_in, const int* in_sizes, int n_in,
                              void* d_out, int out_size, void* d_ws,
                              size_t ws_size, hipStream_t stream) {
  (void)in_sizes; (void)n_in; (void)out_size; (void)ws_size;
  const float* v = (const float*)d_in[0];
  const float* a = (const float*)d_in[1];
  const float* t = (const float*)d_in[2];
  float* ws  = (float*)d_ws;   // needs 7 * N * 4 = 448 KiB
  float* out = (float*)d_out;  // 2 floats

  // 4 tiles (of 16 rows) per block, 2 waves per tile (K split) -> 256 blocks,
  // 2048 waves total for bandwidth saturation.
  nce_pass1<<<dim3(N_ROWS / 16 / 4), dim3(256), 0, stream>>>(v, a, t, ws);
  nce_pass2<<<dim3(1), dim3(1024), 0, stream>>>(v, a, t, ws, out);
}